// WeightedTensorProduct_11536282157112
// MI455X (gfx1250) — compile-verified
//
#include <hip/hip_runtime.h>

typedef __attribute__((ext_vector_type(2))) float v2f;
typedef __attribute__((ext_vector_type(8))) float v8f;

#define D1 52      // irrep dimension (8*1 + 8*3 + 4*5)
#define DD 2704    // 52*52
#define KT 13      // 52 / 4 k-steps for the apply kernel

// ---------------------------------------------------------------------------
// Kernel 0: zero-padded copy of W:  Wp[64, H] <- W[52, H], rows 52..63 = 0.
// Removes every lane-divergent guard from the WMMA loops downstream.
// ---------------------------------------------------------------------------
__global__ void __launch_bounds__(256)
wtp_pad_w(const float* __restrict__ Wm, float* __restrict__ Wp, int H) {
  const int idx = blockIdx.x * 256 + threadIdx.x;
  const int total = 64 * H;
  if (idx < total) {
    const int r = idx / H;
    const int c = idx - r * H;
    Wp[idx] = (r < D1) ? Wm[(size_t)r * H + c] : 0.0f;
  }
}

// ---------------------------------------------------------------------------
// Kernel 1: fold the static matrices once per launch:
//   Ap[64, 2704] = Wp[64, H] @ cb[H, 2704]     (rows >= 52 come out zero)
// One block per 16-wide column tile (169 blocks), 4 waves = 4 row tiles.
// 4 k-steps per iteration: one b64 load clause, then 4 chained WMMAs.
// ---------------------------------------------------------------------------
__global__ void __launch_bounds__(128)
wtp_fold_weights(const float* __restrict__ Wp, const float* __restrict__ cb,
                 float* __restrict__ Ap, int H) {
  const int lane  = threadIdx.x & 31;
  const int wave  = threadIdx.x >> 5;       // 0..3 -> M tile
  const int l15   = lane & 15;
  const int khalf = (lane < 16) ? 0 : 2;    // A/B frag: lane-half selects K+2
  const int n     = blockIdx.x * 16 + l15;  // output column 0..2703
  const int m     = wave * 16 + l15;        // row of Wp, 0..63, always valid

  const float* wrow = Wp + (size_t)m * H + khalf;      // 8B aligned
  const float* cbp  = cb + (size_t)khalf * DD + n;     // column-strided base

  v8f acc = {};
  int kb = 0;
  for (; kb + 16 <= H; kb += 16) {
    v2f af[4], bf[4];
#pragma unroll
    for (int j = 0; j < 4; ++j)
      af[j] = *(const v2f*)(wrow + kb + 4 * j);        // global_load_b64
#pragma unroll
    for (int j = 0; j < 4; ++j) {
      bf[j].x = cbp[(size_t)(kb + 4 * j) * DD];
      bf[j].y = cbp[(size_t)(kb + 4 * j + 1) * DD];
    }
#pragma unroll
    for (int j = 0; j < 4; ++j)
      acc = __builtin_amdgcn_wmma_f32_16x16x4_f32(false, af[j], false, bf[j],
                                                  (short)0, acc, false, false);
  }
  for (; kb < H; kb += 4) {                            // H % 16 remainder
    v2f af = *(const v2f*)(wrow + kb);
    v2f bf;
    bf.x = cbp[(size_t)kb * DD];
    bf.y = cbp[(size_t)(kb + 1) * DD];
    acc = __builtin_amdgcn_wmma_f32_16x16x4_f32(false, af, false, bf,
                                                (short)0, acc, false, false);
  }

  const int hi8 = (lane >= 16) ? 8 : 0;     // D frag: lanes 16-31 hold rows +8
#pragma unroll
  for (int v = 0; v < 8; ++v)
    Ap[(size_t)(wave * 16 + hi8 + v) * DD + n] = acc[v];
}

// ---------------------------------------------------------------------------
// Kernel 2: fused apply.
//   P[(o_l*52+i), n] = sum_j Ap[oc*16+o_l, i*52+j] * in2[bt*16+n, j]   (WMMA)
//   out[b, o]        = sum_i in1[b, i] * P[(o_l*52+i), b_l]            (VALU)
// Block = 256 threads (8 waves) handles a 16-batch x 16-output tile.
// ---------------------------------------------------------------------------
__global__ void __launch_bounds__(256)
wtp_apply(const float* __restrict__ in1, const float* __restrict__ in2,
          const float* __restrict__ Ap, float* __restrict__ out, int B) {
  __shared__ float P[832 * 17];             // 56,576 B; stride 17 pads banks

  const int bt    = blockIdx.x;             // batch tile
  const int oc    = blockIdx.y;             // output chunk 0..3
  const int lane  = threadIdx.x & 31;
  const int wave  = threadIdx.x >> 5;       // 0..7
  const int l15   = lane & 15;
  const int khalf = (lane < 16) ? 0 : 2;
  const int hi8   = (lane >= 16) ? 8 : 0;

  // Prefetch the in1 rows used by the tail contraction (global_prefetch_b8).
  {
    int bp = bt * 16 + (threadIdx.x & 15);
    if (bp >= B) bp = B - 1;
    __builtin_prefetch(in1 + (size_t)bp * D1, 0, 0);
  }

  // Stage all 13 B-fragments of the in2^T tile [K=52, N=16] in registers.
  // Clamp instead of guard: out-of-range lanes duplicate row B-1; their P
  // columns are never stored, so no exec-mask games in the loop.
  const int bcol   = bt * 16 + l15;
  const int bclamp = (bcol < B) ? bcol : (B - 1);
  const float* b2  = in2 + (size_t)bclamp * D1 + khalf;   // 8B aligned
  v2f bfrag[KT];
#pragma unroll
  for (int kt = 0; kt < KT; ++kt)
    bfrag[kt] = *(const v2f*)(b2 + 4 * kt);               // global_load_b64

  // 52 M-tiles of the 832-row (o_l, i) dimension, round-robined over 8 waves.
  for (int mt = wave; mt < 52; mt += 8) {
    // Skip tiles whose outputs are entirely padding (wave-uniform branch).
    if (oc * 16 + (mt * 16) / D1 >= D1) break;

    const int m   = mt * 16 + l15;          // 0..831
    const int o_l = m / D1;
    const int i   = m - o_l * D1;
    // Ap is padded to 64 rows -> always in-bounds, no per-lane guard.
    const float* arow = Ap + (size_t)(oc * 16 + o_l) * DD
                           + (size_t)i * D1 + khalf;      // 8B aligned

    v2f af[KT];
#pragma unroll
    for (int kt = 0; kt < KT; ++kt)
      af[kt] = *(const v2f*)(arow + 4 * kt);              // b64 load clause

    v8f acc = {};
#pragma unroll
    for (int kt = 0; kt < KT; ++kt)
      acc = __builtin_amdgcn_wmma_f32_16x16x4_f32(false, af[kt], false,
                                                  bfrag[kt], (short)0, acc,
                                                  false, false);
#pragma unroll
    for (int v = 0; v < 8; ++v)
      P[(mt * 16 + hi8 + v) * 17 + l15] = acc[v];
  }
  __syncthreads();

  // Contraction with in1: exactly one (o, b) pair per thread (256 = 16x16).
  const int o_l2 = threadIdx.x >> 4;
  const int b_l2 = threadIdx.x & 15;
  const int o    = oc * 16 + o_l2;
  const int b    = bt * 16 + b_l2;
  if (o < D1 && b < B) {
    const float* r1 = in1 + (size_t)b * D1;
    float s = 0.0f;
#pragma unroll 4
    for (int i = 0; i < D1; ++i)
      s = __builtin_fmaf(r1[i], P[(o_l2 * D1 + i) * 17 + b_l2], s);
    out[(size_t)b * D1 + o] = s;
  }
}

// ---------------------------------------------------------------------------
extern "C" void kernel_launch(void* const* d_in, const int* in_sizes, int n_in,
                              void* d_out, int out_size, void* d_ws, size_t ws_size,
                              hipStream_t stream) {
  (void)n_in; (void)out_size; (void)ws_size;
  const float* in1 = (const float*)d_in[0];   // [B, 52]
  const float* in2 = (const float*)d_in[1];   // [B, 52]
  const float* cb  = (const float*)d_in[2];   // [H, 2704]
  const float* Wm  = (const float*)d_in[3];   // [52, H]
  float*       out = (float*)d_out;           // [B, 52]

  const int H = in_sizes[3] / D1;             // 2000
  const int B = in_sizes[0] / D1;             // 4096

  float* Wp = (float*)d_ws;                   // [64, H]    padded weights
  float* Ap = Wp + (size_t)64 * H;            // [64, 2704] folded matrix

  wtp_pad_w<<<(64 * H + 255) / 256, 256, 0, stream>>>(Wm, Wp, H);
  wtp_fold_weights<<<DD / 16, 128, 0, stream>>>(Wp, cb, Ap, H);

  dim3 grid((B + 15) / 16, (D1 + 15) / 16);   // 256 x 4 blocks
  wtp_apply<<<grid, 256, 0, stream>>>(in1, in2, Ap, out, B);
}